// EfficientAttention_58695023067326
// MI455X (gfx1250) — compile-verified
//
#include <hip/hip_runtime.h>

typedef float v2f __attribute__((ext_vector_type(2)));
typedef float v8f __attribute__((ext_vector_type(8)));

#define B_ 4
#define S_ 8192
#define H_ 16
#define D_ 64
#define ROWSTR 3072              // 3*H*D floats between consecutive s
#define BSTR ((size_t)S_ * ROWSTR)
#define NSPLIT 8
#define SPS (S_ / NSPLIT)        // 1024 s-values per split
#define KXS 72                   // LDS stride for k/v tiles (bank-conflict-free frags)
#define QSTR 68                  // LDS stride for q tile
#define CSTR 72                  // LDS stride for context tile

// workspace layout (floats)
#define PMAX_OFF 0               // [NSPLIT][64*64]   = 32768
#define KSUM_OFF 32768           // [64*64]           = 4096
#define CTX_OFF  36864           // [64][64*64]       = 262144
#define WS_FLOATS (CTX_OFF + 64 * 4096)

static __device__ __forceinline__ v8f wmma4(v2f a, v2f b, v8f c) {
  // D = A(16x4 f32) x B(4x16 f32) + C(16x16 f32)   -> v_wmma_f32_16x16x4_f32
  return __builtin_amdgcn_wmma_f32_16x16x4_f32(false, a, false, b, (short)0, c,
                                               false, false);
}

// ---------------------------------------------------------------------------
// Pass 1: partial max over S of k[b,s,h,d] per (split, b, h, d)
// ---------------------------------------------------------------------------
__global__ void ea_kmax_kernel(const float* __restrict__ qkv,
                               float* __restrict__ ws) {
  const int bh = blockIdx.x >> 3;
  const int split = blockIdx.x & 7;
  const int b = bh >> 4, h = bh & 15;
  const int t = threadIdx.x;
  const int d = t & 63;
  const int sg = t >> 6;  // 0..3

  const float* kp = qkv + (size_t)b * BSTR + (size_t)H_ * D_ + h * D_ + d;
  const int s0 = split * SPS;
  float m = -3.4e38f;
  for (int s = sg; s < SPS; s += 4)
    m = fmaxf(m, kp[(size_t)(s0 + s) * ROWSTR]);

  __shared__ float red[256];
  red[t] = m;
  __syncthreads();
  if (t < 64) {
    float mm = fmaxf(fmaxf(red[t], red[t + 64]), fmaxf(red[t + 128], red[t + 192]));
    ws[PMAX_OFF + split * 4096 + bh * 64 + d] = mm;
  }
}

// ---------------------------------------------------------------------------
// Pass 2: partial context[d,e] += sum_s exp(k[s,d]-kmax[d]) * v[s,e]
//         partial ksum[d]      += sum_s exp(k[s,d]-kmax[d])
// One workgroup per (bh, split); fp32 WMMA over the s (K) dimension.
// ---------------------------------------------------------------------------
__global__ void ea_ctx_kernel(const float* __restrict__ qkv,
                              float* __restrict__ ws) {
  __shared__ float kx[64 * KXS];
  __shared__ float vx[64 * KXS];
  __shared__ float kmaxl[64];
  __shared__ float sums[256 * 4];

  const int bh = blockIdx.x >> 3;
  const int split = blockIdx.x & 7;
  const int b = bh >> 4, h = bh & 15;
  const int t = threadIdx.x;
  const int lane = t & 31;
  const int wave = t >> 5;

  // finalize column max across the 8 partials
  if (t < 64) {
    float m = ws[PMAX_OFF + bh * 64 + t];
    #pragma unroll
    for (int sp = 1; sp < NSPLIT; ++sp)
      m = fmaxf(m, ws[PMAX_OFF + sp * 4096 + bh * 64 + t]);
    kmaxl[t] = m;
  }
  __syncthreads();

  const float* kbase = qkv + (size_t)b * BSTR + (size_t)H_ * D_ + h * D_;
  const float* vbase = qkv + (size_t)b * BSTR + (size_t)2 * H_ * D_ + h * D_;

  // tile assignment: 16 output tiles (4x4 grid of 16x16), 2 per wave
  const int t0 = wave * 2, t1 = wave * 2 + 1;
  const int dt0 = (t0 >> 2) * 16, et0 = (t0 & 3) * 16;
  const int dt1 = (t1 >> 2) * 16, et1 = (t1 & 3) * 16;
  const int ml = lane & 15;       // m (or n) index within fragment
  const int hh = lane >> 4;       // lane half -> K offset 2*hh

  const int d0 = 4 * (t & 15);    // load column group
  const int srow = t >> 4;        // load row base (0..15)

  v8f acc0 = {}; v8f acc1 = {};
  float ps0 = 0.f, ps1 = 0.f, ps2 = 0.f, ps3 = 0.f;

  for (int ch = 0; ch < SPS / 64; ++ch) {
    const int s0 = split * SPS + ch * 64;
    // cooperative load of 64x64 chunk of k (exp'ed) and v
    #pragma unroll
    for (int j = 0; j < 4; ++j) {
      const int sl = srow + j * 16;
      const float4 k4 = *(const float4*)(kbase + (size_t)(s0 + sl) * ROWSTR + d0);
      float4 e4;
      e4.x = __expf(k4.x - kmaxl[d0 + 0]);
      e4.y = __expf(k4.y - kmaxl[d0 + 1]);
      e4.z = __expf(k4.z - kmaxl[d0 + 2]);
      e4.w = __expf(k4.w - kmaxl[d0 + 3]);
      ps0 += e4.x; ps1 += e4.y; ps2 += e4.z; ps3 += e4.w;
      *(float4*)(kx + sl * KXS + d0) = e4;
      const float4 v4 = *(const float4*)(vbase + (size_t)(s0 + sl) * ROWSTR + d0);
      *(float4*)(vx + sl * KXS + d0) = v4;
    }
    __syncthreads();

    // context tiles: K-loop over the 64 s values of this chunk, 4 at a time
    #pragma unroll
    for (int kk0 = 0; kk0 < 64; kk0 += 4) {
      const int sA = kk0 + 2 * hh;
      v2f a0, b0, a1, b1;
      a0.x = kx[(sA + 0) * KXS + dt0 + ml];
      a0.y = kx[(sA + 1) * KXS + dt0 + ml];
      b0.x = vx[(sA + 0) * KXS + et0 + ml];
      b0.y = vx[(sA + 1) * KXS + et0 + ml];
      acc0 = wmma4(a0, b0, acc0);
      a1.x = kx[(sA + 0) * KXS + dt1 + ml];
      a1.y = kx[(sA + 1) * KXS + dt1 + ml];
      b1.x = vx[(sA + 0) * KXS + et1 + ml];
      b1.y = vx[(sA + 1) * KXS + et1 + ml];
      acc1 = wmma4(a1, b1, acc1);
    }
    __syncthreads();
  }

  // accumulate partial context into global workspace
  float* ctx = ws + CTX_OFF + (size_t)bh * 4096;
  #pragma unroll
  for (int r = 0; r < 8; ++r) {
    const int dr = r + 8 * hh;  // C/D row: vgpr r -> M=r (lanes 0-15) / r+8 (16-31)
    atomicAdd(ctx + (dt0 + dr) * 64 + et0 + ml, acc0[r]);
    atomicAdd(ctx + (dt1 + dr) * 64 + et1 + ml, acc1[r]);
  }

  // accumulate partial softmax denominators
  sums[t * 4 + 0] = ps0; sums[t * 4 + 1] = ps1;
  sums[t * 4 + 2] = ps2; sums[t * 4 + 3] = ps3;
  __syncthreads();
  if (t < 16) {
    float s0a = 0.f, s1a = 0.f, s2a = 0.f, s3a = 0.f;
    #pragma unroll
    for (int mmb = 0; mmb < 16; ++mmb) {
      const int idx = (t + 16 * mmb) * 4;
      s0a += sums[idx + 0]; s1a += sums[idx + 1];
      s2a += sums[idx + 2]; s3a += sums[idx + 3];
    }
    atomicAdd(ws + KSUM_OFF + bh * 64 + 4 * t + 0, s0a);
    atomicAdd(ws + KSUM_OFF + bh * 64 + 4 * t + 1, s1a);
    atomicAdd(ws + KSUM_OFF + bh * 64 + 4 * t + 2, s2a);
    atomicAdd(ws + KSUM_OFF + bh * 64 + 4 * t + 3, s3a);
  }
}

// ---------------------------------------------------------------------------
// Pass 3: out[s,e] = softmax_D(q[s,:]) @ (context / ksum[:,None])
// One workgroup per (bh, 128-row block); each wave owns 16 rows, 4 WMMA tiles.
// ---------------------------------------------------------------------------
__global__ void ea_out_kernel(const float* __restrict__ qkv,
                              const float* __restrict__ ws,
                              float* __restrict__ out) {
  __shared__ float ctx[64 * CSTR];
  __shared__ float hq[128 * QSTR];
  __shared__ float kinv[64];
  __shared__ float rinv[128];

  const int bh = blockIdx.x >> 6;
  const int rb = blockIdx.x & 63;
  const int b = bh >> 4, h = bh & 15;
  const int t = threadIdx.x;
  const int lane = t & 31;
  const int wave = t >> 5;

  if (t < 64) kinv[t] = 1.0f / ws[KSUM_OFF + bh * 64 + t];
  __syncthreads();

  // load + row-scale the 64x64 context into LDS (L2-resident)
  const float* ctxg = ws + CTX_OFF + (size_t)bh * 4096;
  #pragma unroll
  for (int j = 0; j < 4; ++j) {
    const int f = t * 4 + j * 1024;
    const int dd = f >> 6, e = f & 63;
    float4 c4 = *(const float4*)(ctxg + f);
    const float sc = kinv[dd];
    c4.x *= sc; c4.y *= sc; c4.z *= sc; c4.w *= sc;
    *(float4*)(ctx + dd * CSTR + e) = c4;
  }

  // load this wave's 16 q rows (coalesced) into padded LDS
  const float* qbase = qkv + (size_t)b * BSTR + (size_t)h * D_;
  const int sbase = rb * 128 + wave * 16;
  #pragma unroll
  for (int it = 0; it < 8; ++it) {
    const int row = 2 * it + (lane >> 4);
    const int col = (lane & 15) * 4;
    const float4 q4 = *(const float4*)(qbase + (size_t)(sbase + row) * ROWSTR + col);
    *(float4*)(hq + (wave * 16 + row) * QSTR + col) = q4;
  }
  __syncthreads();

  // per-row softmax over D=64: lane pair (L, L^16) splits the row
  {
    const int m = lane & 15, hh = lane >> 4;
    float* rowp = hq + (wave * 16 + m) * QSTR + hh * 32;
    float mx = rowp[0];
    #pragma unroll
    for (int i = 1; i < 32; ++i) mx = fmaxf(mx, rowp[i]);
    mx = fmaxf(mx, __shfl_xor(mx, 16, 32));
    float ssum = 0.f;
    #pragma unroll
    for (int i = 0; i < 32; ++i) {
      const float e = __expf(rowp[i] - mx);
      rowp[i] = e;
      ssum += e;
    }
    ssum += __shfl_xor(ssum, 16, 32);
    if (hh == 0) rinv[wave * 16 + m] = 1.0f / ssum;
  }
  // wave-private LDS regions; DS ops from the same wave are in-order

  const int ml = lane & 15;
  const int hh = lane >> 4;
  v8f acc0 = {}, acc1 = {}, acc2 = {}, acc3 = {};
  #pragma unroll
  for (int kk0 = 0; kk0 < 64; kk0 += 4) {
    const int kA = kk0 + 2 * hh;
    v2f a;
    a.x = hq[(wave * 16 + ml) * QSTR + kA + 0];
    a.y = hq[(wave * 16 + ml) * QSTR + kA + 1];
    v2f b0, b1, b2, b3;
    b0.x = ctx[(kA + 0) * CSTR + 0 + ml];  b0.y = ctx[(kA + 1) * CSTR + 0 + ml];
    b1.x = ctx[(kA + 0) * CSTR + 16 + ml]; b1.y = ctx[(kA + 1) * CSTR + 16 + ml];
    b2.x = ctx[(kA + 0) * CSTR + 32 + ml]; b2.y = ctx[(kA + 1) * CSTR + 32 + ml];
    b3.x = ctx[(kA + 0) * CSTR + 48 + ml]; b3.y = ctx[(kA + 1) * CSTR + 48 + ml];
    acc0 = wmma4(a, b0, acc0);
    acc1 = wmma4(a, b1, acc1);
    acc2 = wmma4(a, b2, acc2);
    acc3 = wmma4(a, b3, acc3);
  }

  // write out, scaling each row by its softmax reciprocal
  #pragma unroll
  for (int r = 0; r < 8; ++r) {
    const int mr = r + 8 * hh;
    const int s = sbase + mr;
    const float ri = rinv[wave * 16 + mr];
    const size_t o = ((size_t)b * S_ + s) * (H_ * D_) + (size_t)h * D_;
    out[o + 0 + ml]  = acc0[r] * ri;
    out[o + 16 + ml] = acc1[r] * ri;
    out[o + 32 + ml] = acc2[r] * ri;
    out[o + 48 + ml] = acc3[r] * ri;
  }
}

// ---------------------------------------------------------------------------
extern "C" void kernel_launch(void* const* d_in, const int* in_sizes, int n_in,
                              void* d_out, int out_size, void* d_ws, size_t ws_size,
                              hipStream_t stream) {
  (void)in_sizes; (void)n_in; (void)out_size; (void)ws_size;
  const float* qkv = (const float*)d_in[0];
  float* out = (float*)d_out;
  float* ws = (float*)d_ws;

  // zero the accumulated regions (ksum + context) every launch
  hipMemsetAsync((char*)d_ws + (size_t)KSUM_OFF * sizeof(float), 0,
                 (size_t)(WS_FLOATS - KSUM_OFF) * sizeof(float), stream);

  ea_kmax_kernel<<<64 * NSPLIT, 256, 0, stream>>>(qkv, ws);
  ea_ctx_kernel<<<64 * NSPLIT, 256, 0, stream>>>(qkv, ws);
  ea_out_kernel<<<64 * (S_ / 128), 256, 0, stream>>>(qkv, ws, out);
}